// MCRGCN_11003706212539
// MI455X (gfx1250) — compile-verified
//
#include <hip/hip_runtime.h>
#include <hip/hip_bf16.h>

typedef float v2f __attribute__((ext_vector_type(2)));
typedef float v8f __attribute__((ext_vector_type(8)));

#define NREL 3
#define PDROP 0.3f
#define BNEPS 1e-5f

// ---------------------------------------------------------------------------
// WMMA GEMM: Out[M,HN] (+)= X[M,K] @ W[K,HN]   (fp32 via V_WMMA_F32_16X16X4_F32)
// Block = 128 threads = 4 waves. One block handles 4 consecutive 16-row M-tiles
// of a single 16-col N-tile. The K x 16 B-panel (<= 4KB) is staged once into
// LDS; each wave then runs the fully-unrolled K/4 WMMA chain reading B frags
// from LDS (ds_load) and A frags from global (v2f, 8B-aligned).
// A frag (16x4 f32): lanes 0-15 hold K=0,1 in v[0],v[1]; lanes 16-31 hold K=2,3.
// C/D frag (16x16 f32): c[i] at lane L -> M = i + 8*(L>>4), N = L&15.
// ---------------------------------------------------------------------------
template <int K, int HN>
__global__ void wmma_gemm(const float* __restrict__ X, const float* __restrict__ W,
                          float* __restrict__ Out, int M, int accumulate)
{
    constexpr int NT = HN >> 4;
    __shared__ float Bs[K * 16];

    const int lane = threadIdx.x & 31;
    const int wave = threadIdx.x >> 5;
    const int mtiles = M >> 4;                    // 100000/16 = 6250
    const int nt = blockIdx.x % NT;
    const int mtBase = (blockIdx.x / NT) * 4;
    const int n0 = nt << 4;

    // ---- stage B panel [K x 16] into LDS (cooperative, coalesced) ----
    for (int idx = threadIdx.x; idx < K * 16; idx += blockDim.x) {
        const int kk = idx >> 4, cc = idx & 15;
        Bs[idx] = W[(long)kk * HN + n0 + cc];
    }
    __syncthreads();

    const int mt = mtBase + wave;                 // wave-uniform
    if (mt >= mtiles) return;                     // EXEC stays full per wave
    const int m0 = mt << 4;
    const int row = lane & 15;
    const int half = lane >> 4;

    v8f c = {};
    if (accumulate) {
#pragma unroll
        for (int i = 0; i < 8; ++i)
            c[i] = Out[(long)(m0 + i + 8 * half) * HN + n0 + row];
    }

    const float* xrow = X + (long)(m0 + row) * K;
#pragma unroll
    for (int k = 0; k < K; k += 4) {
        v2f a = *(const v2f*)(xrow + k + 2 * half);        // 8B aligned
        v2f b;
        b.x = Bs[(k + 2 * half) * 16 + row];               // ds_load, conflict-free
        b.y = Bs[(k + 2 * half + 1) * 16 + row];
        c = __builtin_amdgcn_wmma_f32_16x16x4_f32(
                /*neg_a=*/false, a, /*neg_b=*/false, b,
                /*c_mod=*/(short)0, c, /*reuse_a=*/false, /*reuse_b=*/false);
    }

#pragma unroll
    for (int i = 0; i < 8; ++i)
        Out[(long)(m0 + i + 8 * half) * HN + n0 + row] = c[i];
}

// ---------------------------------------------------------------------------
// Out[n*H + f] = bias[f]
// ---------------------------------------------------------------------------
__global__ void bias_init(float* __restrict__ Out, const float* __restrict__ bias,
                          long total, int H)
{
    long i = (long)blockIdx.x * blockDim.x + threadIdx.x;
    if (i < total) Out[i] = bias[(int)(i % H)];
}

// ---------------------------------------------------------------------------
// cnt[r*N + dst] += 1 per edge
// ---------------------------------------------------------------------------
__global__ void count_edges(const int* __restrict__ dst, const int* __restrict__ et,
                            float* __restrict__ cnt, long E, long N)
{
    long e = (long)blockIdx.x * blockDim.x + threadIdx.x;
    if (e >= E) return;
    atomicAdd(&cnt[(long)et[e] * N + dst[e]], 1.0f);
}

__global__ void inv_counts(float* __restrict__ cnt, long total)
{
    long i = (long)blockIdx.x * blockDim.x + threadIdx.x;
    if (i < total) cnt[i] = 1.0f / fmaxf(cnt[i], 1.0f);
}

// ---------------------------------------------------------------------------
// For edges of relation r: Out[dst, f] += T[src, f] * inv[r*N + dst]
// (mean aggregation folded into precomputed 1/deg)
// ---------------------------------------------------------------------------
__global__ void scatter_rel(const int* __restrict__ src, const int* __restrict__ dst,
                            const int* __restrict__ et, const float* __restrict__ T,
                            const float* __restrict__ inv, float* __restrict__ Out,
                            long E, int H, int r, long N)
{
    const int epb = blockDim.x / H;
    long e = (long)blockIdx.x * epb + (threadIdx.x / H);
    const int f = threadIdx.x % H;
    if (e >= E) return;
    if (e + 2048 < E) __builtin_prefetch(&et[e + 2048], 0, 1);  // global_prefetch_b8
    if (et[e] != r) return;
    const int s = src[e], d = dst[e];
    atomicAdd(&Out[(long)d * H + f], T[(long)s * H + f] * inv[(long)r * N + d]);
}

// ---------------------------------------------------------------------------
// BatchNorm stats: stats[f] = sum, stats[H+f] = sumsq  (atomics across blocks)
// ---------------------------------------------------------------------------
__global__ void bn_stats(const float* __restrict__ Hb, float* __restrict__ stats,
                         int H, long N)
{
    const int rpb = blockDim.x / H;
    const int f = threadIdx.x % H;
    long r0 = (long)blockIdx.x * rpb + (threadIdx.x / H);
    long stride = (long)gridDim.x * rpb;
    float s = 0.0f, s2 = 0.0f;
    for (long r = r0; r < N; r += stride) {
        float v = Hb[r * H + f];
        s += v; s2 += v * v;
    }
    atomicAdd(&stats[f], s);
    atomicAdd(&stats[H + f], s2);
}

__device__ __forceinline__ float hash_uniform(unsigned long i, unsigned seed)
{
    unsigned x = (unsigned)i * 2654435761u ^ (seed * 0x9E3779B9u);
    x ^= x >> 16; x *= 0x7feb352du; x ^= x >> 15; x *= 0x846ca68bu; x ^= x >> 16;
    return (float)(x >> 8) * (1.0f / 16777216.0f);
}

// ---------------------------------------------------------------------------
// In-place: BN -> ReLU -> dropout
// ---------------------------------------------------------------------------
__global__ void bn_relu_drop(float* __restrict__ Hb, const float* __restrict__ stats,
                             const float* __restrict__ g, const float* __restrict__ be,
                             long N, int H, unsigned seed)
{
    long i = (long)blockIdx.x * blockDim.x + threadIdx.x;
    if (i >= N * (long)H) return;
    const int f = (int)(i % H);
    const float mean = stats[f] / (float)N;
    const float var  = stats[H + f] / (float)N - mean * mean;
    float v = (Hb[i] - mean) * rsqrtf(var + BNEPS) * g[f] + be[f];
    v = fmaxf(v, 0.0f);
    const float u = hash_uniform((unsigned long)i, seed);
    Hb[i] = (u >= PDROP) ? v * (1.0f / (1.0f - PDROP)) : 0.0f;
}

// ---------------------------------------------------------------------------
// In-place: ReLU -> dropout (classifier hidden)
// ---------------------------------------------------------------------------
__global__ void relu_drop(float* __restrict__ Hb, long total, unsigned seed)
{
    long i = (long)blockIdx.x * blockDim.x + threadIdx.x;
    if (i >= total) return;
    float v = fmaxf(Hb[i], 0.0f);
    const float u = hash_uniform((unsigned long)i, seed);
    Hb[i] = (u >= PDROP) ? v * (1.0f / (1.0f - PDROP)) : 0.0f;
}

// ---------------------------------------------------------------------------
// Final head: out[n,0:2] = h3[n,:] @ Wc2[32,2] + bc2
// ---------------------------------------------------------------------------
__global__ void classifier_out(const float* __restrict__ H3, const float* __restrict__ Wc2,
                               const float* __restrict__ bc2, float* __restrict__ out,
                               long N)
{
    long n = (long)blockIdx.x * blockDim.x + threadIdx.x;
    if (n >= N) return;
    float a0 = bc2[0], a1 = bc2[1];
    const float* hr = H3 + n * 32;
#pragma unroll
    for (int k = 0; k < 32; ++k) {
        const float h = hr[k];
        a0 += h * Wc2[k * 2 + 0];
        a1 += h * Wc2[k * 2 + 1];
    }
    out[n * 2 + 0] = a0;
    out[n * 2 + 1] = a1;
}

// ---------------------------------------------------------------------------
static inline size_t al256(size_t x) { return (x + 255) & ~(size_t)255; }

extern "C" void kernel_launch(void* const* d_in, const int* in_sizes, int n_in,
                              void* d_out, int out_size, void* d_ws, size_t ws_size,
                              hipStream_t stream)
{
    const float* x      = (const float*)d_in[0];
    const int*   ei     = (const int*)  d_in[1];
    const int*   et     = (const int*)  d_in[2];
    const float* Wself1 = (const float*)d_in[3];
    const float* Wrel1  = (const float*)d_in[4];
    const float* b1     = (const float*)d_in[5];
    const float* g1     = (const float*)d_in[6];
    const float* be1    = (const float*)d_in[7];
    const float* Wself2 = (const float*)d_in[8];
    const float* Wrel2  = (const float*)d_in[9];
    const float* b2     = (const float*)d_in[10];
    const float* g2     = (const float*)d_in[11];
    const float* be2    = (const float*)d_in[12];
    const float* Wc1    = (const float*)d_in[13];
    const float* bc1    = (const float*)d_in[14];
    const float* Wc2    = (const float*)d_in[15];
    const float* bc2    = (const float*)d_in[16];

    const long N = in_sizes[0] / 64;      // 100000
    const long E = in_sizes[1] / 2;       // 1200000
    const int* srcI = ei;
    const int* dstI = ei + E;

    // workspace partition
    char* w = (char*)d_ws;
    float* cnt   = (float*)w; w += al256((size_t)NREL * N * 4);  // then holds 1/deg
    float* stats = (float*)w; w += 256;                          // up to 128 floats
    float* T     = (float*)w; w += al256((size_t)N * 64 * 4);
    float* h1    = (float*)w; w += al256((size_t)N * 64 * 4);
    float* h2    = (float*)w; w += al256((size_t)N * 32 * 4);
    float* h3    = (float*)w; w += al256((size_t)N * 32 * 4);

    const int B = 256;
    const int GEMM_B = 128;               // 4 waves: 4 M-tiles sharing one N-tile
    const long mtiles = N / 16;
    auto gemm_grid = [&](int ntiles) {
        return (int)(((mtiles + 3) / 4) * ntiles);
    };

    // ---- per-(relation,dst) degree -> reciprocal -----------------------------
    hipMemsetAsync(cnt, 0, (size_t)NREL * N * 4, stream);
    count_edges<<<(int)((E + B - 1) / B), B, 0, stream>>>(dstI, et, cnt, E, N);
    inv_counts<<<(int)((NREL * N + B - 1) / B), B, 0, stream>>>(cnt, NREL * N);

    // ---- layer 1: h1 = b1 + x@Wself1 + sum_r mean_r(x@Wrel1[r]) --------------
    bias_init<<<(int)((N * 64 + B - 1) / B), B, 0, stream>>>(h1, b1, N * 64, 64);
    wmma_gemm<64, 64><<<gemm_grid(4), GEMM_B, 0, stream>>>(x, Wself1, h1, (int)N, 1);
    for (int r = 0; r < NREL; ++r) {
        wmma_gemm<64, 64><<<gemm_grid(4), GEMM_B, 0, stream>>>(
            x, Wrel1 + (size_t)r * 64 * 64, T, (int)N, 0);
        scatter_rel<<<(int)((E + 3) / 4), B, 0, stream>>>(srcI, dstI, et, T, cnt, h1,
                                                          E, 64, r, N);
    }
    hipMemsetAsync(stats, 0, 2 * 64 * 4, stream);
    bn_stats<<<1024, B, 0, stream>>>(h1, stats, 64, N);
    bn_relu_drop<<<(int)((N * 64 + B - 1) / B), B, 0, stream>>>(h1, stats, g1, be1, N, 64, 1u);

    // ---- layer 2: h2 = b2 + h1@Wself2 + sum_r mean_r(h1@Wrel2[r]) ------------
    bias_init<<<(int)((N * 32 + B - 1) / B), B, 0, stream>>>(h2, b2, N * 32, 32);
    wmma_gemm<64, 32><<<gemm_grid(2), GEMM_B, 0, stream>>>(h1, Wself2, h2, (int)N, 1);
    for (int r = 0; r < NREL; ++r) {
        wmma_gemm<64, 32><<<gemm_grid(2), GEMM_B, 0, stream>>>(
            h1, Wrel2 + (size_t)r * 64 * 32, T, (int)N, 0);
        scatter_rel<<<(int)((E + 7) / 8), B, 0, stream>>>(srcI, dstI, et, T, cnt, h2,
                                                          E, 32, r, N);
    }
    hipMemsetAsync(stats, 0, 2 * 32 * 4, stream);
    bn_stats<<<1024, B, 0, stream>>>(h2, stats, 32, N);
    bn_relu_drop<<<(int)((N * 32 + B - 1) / B), B, 0, stream>>>(h2, stats, g2, be2, N, 32, 2u);

    // ---- classifier ----------------------------------------------------------
    bias_init<<<(int)((N * 32 + B - 1) / B), B, 0, stream>>>(h3, bc1, N * 32, 32);
    wmma_gemm<32, 32><<<gemm_grid(2), GEMM_B, 0, stream>>>(h2, Wc1, h3, (int)N, 1);
    relu_drop<<<(int)((N * 32 + B - 1) / B), B, 0, stream>>>(h3, N * 32, 3u);
    classifier_out<<<(int)((N + B - 1) / B), B, 0, stream>>>(h3, Wc2, bc2, (float*)d_out, N);

    (void)n_in; (void)out_size; (void)ws_size;
}